// CholeskyBatchNormSPD_12584254177384
// MI455X (gfx1250) — compile-verified
//
#include <hip/hip_runtime.h>
#include <math.h>

typedef float v2f __attribute__((ext_vector_type(2)));
typedef float v8f __attribute__((ext_vector_type(8)));

#define REC      1025                 // 1024 coord sums + 1 sumsq per block record
#define NB1      128                  // blocks in stats pass
#define MEAN_OFF (NB1 * REC)          // floats
#define FACT_OFF (MEAN_OFF + 1024)    // floats

// ---- helpers ---------------------------------------------------------------

// Uniform broadcast from a compile-time-constant lane: lowers to v_readlane_b32
// (SGPR result) instead of ds_bpermute + s_wait_dscnt.
__device__ __forceinline__ float bcast(float v, int lane) {
  return __int_as_float(__builtin_amdgcn_readlane(__float_as_int(v), lane));
}

__device__ __forceinline__ void load_rows(const float* __restrict__ p, float r[32]) {
#pragma unroll
  for (int q = 0; q < 8; ++q) {
    float4 v = reinterpret_cast<const float4*>(p)[q];
    r[4*q+0] = v.x; r[4*q+1] = v.y; r[4*q+2] = v.z; r[4*q+3] = v.w;
  }
}

// In-register right-looking Cholesky of a 32x32 SPD matrix; lane i holds row i.
// On return r[j] (j <= lane) is L[lane][j]; returns L[lane][lane].
// All cross-lane traffic is constant-lane readlane broadcasts (no LDS pipe).
__device__ __forceinline__ float cholesky32(float r[32], int lane) {
  float diag = 1.0f;
#pragma unroll
  for (int k = 0; k < 32; ++k) {
    float dk  = bcast(r[k], k);               // current (k,k), uniform SGPR
    float inv = 1.0f / sqrtf(dk);
    r[k] *= inv;                               // column k -> L[:,k]
    if (lane == k) diag = r[k];                // sqrt(dk) = L[k][k]
#pragma unroll
    for (int j = k + 1; j < 32; ++j) {
      r[j] = fmaf(-r[k], bcast(r[k], j), r[j]);   // trailing update, scalar operand
    }
  }
  return diag;
}

// ---- pass 1: Cholesky + Log-Cholesky coordinate sums -----------------------

extern "C" __global__ void __launch_bounds__(256)
k1_chol_stats(const float* __restrict__ X, float* __restrict__ ws, int nmat) {
  __shared__ float lds[8 * 1024];
  __shared__ float ldss[8];
  const int lane = threadIdx.x & 31;
  const int wv   = threadIdx.x >> 5;
  const int gw   = blockIdx.x * 8 + wv;
  const int nw   = gridDim.x * 8;

  float acc[32];
#pragma unroll
  for (int j = 0; j < 32; ++j) acc[j] = 0.0f;
  float sumsq = 0.0f;

  for (int m = gw; m < nmat; m += nw) {
    float r[32];
    load_rows(X + (size_t)m * 1024 + lane * 32, r);
    float diag = cholesky32(r, lane);
    float logd = logf(diag);
#pragma unroll
    for (int j = 0; j < 32; ++j) {
      float c = (j < lane) ? r[j] : 0.0f;   // strictly-lower coord
      if (j == lane) c = logd;              // log-diag coord
      acc[j] += c;
      sumsq = fmaf(c, c, sumsq);
    }
  }

#pragma unroll
  for (int j = 0; j < 32; ++j) lds[wv * 1024 + lane * 32 + j] = acc[j];
#pragma unroll
  for (int off = 16; off > 0; off >>= 1) sumsq += __shfl_xor(sumsq, off, 32);
  if (lane == 0) ldss[wv] = sumsq;
  __syncthreads();

  float* rec = ws + (size_t)blockIdx.x * REC;
#pragma unroll
  for (int q = 0; q < 4; ++q) {
    int p = threadIdx.x + 256 * q;
    float s = 0.0f;
#pragma unroll
    for (int w = 0; w < 8; ++w) s += lds[w * 1024 + p];
    rec[p] = s;
  }
  if (threadIdx.x == 0) {
    float s = 0.0f;
    for (int w = 0; w < 8; ++w) s += ldss[w];
    rec[1024] = s;
  }
}

// ---- pass 2: finalize mean / var / factor ----------------------------------

extern "C" __global__ void __launch_bounds__(256)
k2_finalize(float* __restrict__ ws, const float* __restrict__ sptr, int nmat) {
  __shared__ float red[256];
  const int tid = threadIdx.x;
  const float invM = 1.0f / (float)nmat;

  float l2 = 0.0f;
#pragma unroll
  for (int q = 0; q < 4; ++q) {
    int p = tid + 256 * q;
    float s = 0.0f;
    for (int b = 0; b < NB1; ++b) s += ws[(size_t)b * REC + p];
    float mean = s * invM;
    ws[MEAN_OFF + p] = mean;
    l2 = fmaf(mean, mean, l2);
  }
  float ssq = (tid < NB1) ? ws[(size_t)tid * REC + 1024] : 0.0f;

  red[tid] = l2; __syncthreads();
  for (int off = 128; off > 0; off >>= 1) { if (tid < off) red[tid] += red[tid + off]; __syncthreads(); }
  float norm2 = red[0]; __syncthreads();

  red[tid] = ssq; __syncthreads();
  for (int off = 128; off > 0; off >>= 1) { if (tid < off) red[tid] += red[tid + off]; __syncthreads(); }

  if (tid == 0) {
    float var = red[0] * invM - norm2;      // E||c||^2 - ||mean||^2
    ws[FACT_OFF] = sptr[0] / sqrtf(var);
  }
}

// ---- pass 3: re-factor, center/scale, reconstruct via fp32 WMMA ------------

extern "C" __global__ void __launch_bounds__(256)
k3_reconstruct(const float* __restrict__ X, const float* __restrict__ ws,
               float* __restrict__ out, int nmat) {
  const int lane = threadIdx.x & 31;
  const int wv   = threadIdx.x >> 5;
  const int m    = blockIdx.x * 8 + wv;
  if (m >= nmat) return;                     // wave-uniform: EXEC stays all-ones

  float r[32];
  load_rows(X + (size_t)m * 1024 + lane * 32, r);
  float diag = cholesky32(r, lane);
  float logd = logf(diag);

  const float factor = ws[FACT_OFF];
  const float* mrow  = ws + MEAN_OFF + lane * 32;

  // r <- L_out rows: factor*(sl - mean) off-diag, exp(factor*(logd - mean)) diag, 0 above
#pragma unroll
  for (int q = 0; q < 8; ++q) {
    float4 mv = reinterpret_cast<const float4*>(mrow)[q];
    float mvj[4] = { mv.x, mv.y, mv.z, mv.w };
#pragma unroll
    for (int t = 0; t < 4; ++t) {
      int j = 4 * q + t;
      float c = (j < lane) ? (r[j] - mvj[t]) * factor : 0.0f;
      if (j == lane) c = expf(factor * (logd - mvj[t]));
      r[j] = c;
    }
  }

  // Gather WMMA 16x4 fp32 operand tiles T[mb][kb] from the row-distributed L_out.
  // Per-lane-varying source lane -> ds_bpermute is the right instruction here.
  // Layout: VGPR0 = {K=4kb+0 (lanes 0-15), K=4kb+2 (lanes 16-31)}, VGPR1 = {+1, +3}.
  // Since C = L*L^T, B-tile(kb, nb) == A-tile(nb, kb): one set of tiles serves both.
  v2f T[2][8];
  const bool hi = lane >= 16;
  const int  sb = lane & 15;
#pragma unroll
  for (int mb = 0; mb < 2; ++mb) {
#pragma unroll
    for (int kb = 0; kb < 8; ++kb) {
      int src = 16 * mb + sb;
      float s0 = __shfl(r[4*kb+0], src, 32);
      float s1 = __shfl(r[4*kb+1], src, 32);
      float s2 = __shfl(r[4*kb+2], src, 32);
      float s3 = __shfl(r[4*kb+3], src, 32);
      v2f t;
      t.x = hi ? s2 : s0;
      t.y = hi ? s3 : s1;
      T[mb][kb] = t;
    }
  }

  float* ob = out + (size_t)m * 1024;
  const int rbase = hi ? 8 : 0;
#pragma unroll
  for (int mb = 0; mb < 2; ++mb) {
#pragma unroll
    for (int nb = 0; nb < 2; ++nb) {
      v8f c = {};
#pragma unroll
      for (int kb = 0; kb < 8; ++kb) {
        // D = A(16x4) * B(4x16) + C, fp32 throughout
        c = __builtin_amdgcn_wmma_f32_16x16x4_f32(
              /*neg_a=*/false, T[mb][kb],
              /*neg_b=*/false, T[nb][kb],
              /*c_mod=*/(short)0, c,
              /*reuse_a=*/false, /*reuse_b=*/false);
      }
      // C/D layout: VGPR v -> row 16*mb + v + (hi?8:0), col 16*nb + (lane&15)
      float* p = ob + (16 * mb + rbase) * 32 + 16 * nb + sb;
#pragma unroll
      for (int v = 0; v < 8; ++v) p[v * 32] = c[v];
    }
  }
}

// ---- launcher ---------------------------------------------------------------

extern "C" void kernel_launch(void* const* d_in, const int* in_sizes, int n_in,
                              void* d_out, int out_size, void* d_ws, size_t ws_size,
                              hipStream_t stream) {
  const float* X = (const float*)d_in[0];
  const float* s = (const float*)d_in[1];
  float* out = (float*)d_out;
  float* ws  = (float*)d_ws;
  const int nmat = in_sizes[0] / 1024;   // 512*64 matrices of 32x32

  k1_chol_stats<<<NB1, 256, 0, stream>>>(X, ws, nmat);
  k2_finalize  <<<1,   256, 0, stream>>>(ws, s, nmat);
  k3_reconstruct<<<(nmat + 7) / 8, 256, 0, stream>>>(X, ws, out, nmat);
}